// LinearDML_15118284882378
// MI455X (gfx1250) — compile-verified
//
#include <hip/hip_runtime.h>
#include <stdint.h>

typedef __attribute__((ext_vector_type(2))) float v2f;
typedef __attribute__((ext_vector_type(8))) float v8f;

namespace {
constexpr int kB   = 64;    // batches
constexpr int kN   = 512;   // sequence length
constexpr int kD   = 32;    // feature dim
constexpr int kC   = 34;    // feature dim + t + y
constexpr int kBlk = 16;    // Woodbury block (WMMA tile M/N)
constexpr int kNB  = kN / kBlk;
constexpr int kRow = kBlk * kC;          // 544 floats = 2176 B per block
}
#define RIDGE_EPS 1e-4f

// ---------------------------------------------------------------------------
// WMMA f32 16x16x4 helpers, operands staged row-major in LDS.
// A layout (ISA 7.12.2, 32-bit A 16x4): lane l, m = l&15, half = l>>4,
//   a[v] = A[m][2*half + v].
// B (4x16) mirrors the K split: b[v] = B[2*half + v][l&15].
// C/D (16x16, 8 VGPRs): c[v] = C[v + 8*half][l&15].
// ---------------------------------------------------------------------------
static __device__ __forceinline__ v8f wmma4(const float* A, int lda,
                                            const float* B, int ldb, v8f c) {
  const int l = threadIdx.x & 31;
  const int half = l >> 4, lr = l & 15;
  v2f a, b;
  a.x = A[lr * lda + 2 * half + 0];
  a.y = A[lr * lda + 2 * half + 1];
  b.x = B[(2 * half + 0) * ldb + lr];
  b.y = B[(2 * half + 1) * ldb + lr];
  return __builtin_amdgcn_wmma_f32_16x16x4_f32(false, a, false, b, (short)0, c,
                                               false, false);
}

// B supplied transposed: B[k][n] = Bt[n][k]
static __device__ __forceinline__ v8f wmma4_bt(const float* A, int lda,
                                               const float* Bt, int ldb, v8f c) {
  const int l = threadIdx.x & 31;
  const int half = l >> 4, lr = l & 15;
  v2f a, b;
  a.x = A[lr * lda + 2 * half + 0];
  a.y = A[lr * lda + 2 * half + 1];
  b.x = Bt[lr * ldb + 2 * half + 0];
  b.y = Bt[lr * ldb + 2 * half + 1];
  return __builtin_amdgcn_wmma_f32_16x16x4_f32(false, a, false, b, (short)0, c,
                                               false, false);
}

// A supplied transposed: A[m][k] = At[k][m]
static __device__ __forceinline__ v8f wmma4_at(const float* At, int lda,
                                               const float* B, int ldb, v8f c) {
  const int l = threadIdx.x & 31;
  const int half = l >> 4, lr = l & 15;
  v2f a, b;
  a.x = At[(2 * half + 0) * lda + lr];
  a.y = At[(2 * half + 1) * lda + lr];
  b.x = B[(2 * half + 0) * ldb + lr];
  b.y = B[(2 * half + 1) * ldb + lr];
  return __builtin_amdgcn_wmma_f32_16x16x4_f32(false, a, false, b, (short)0, c,
                                               false, false);
}

static __device__ __forceinline__ void store_tile(float* D, int ldd, v8f c,
                                                  float scale) {
  const int l = threadIdx.x & 31;
  const int half = l >> 4, col = l & 15;
#pragma unroll
  for (int v = 0; v < 8; ++v) D[(v + 8 * half) * ldd + col] = scale * c[v];
}

static __device__ __forceinline__ v8f load_tile(const float* C, int ldc) {
  const int l = threadIdx.x & 31;
  const int half = l >> 4, col = l & 15;
  v8f c;
#pragma unroll
  for (int v = 0; v < 8; ++v) c[v] = C[(v + 8 * half) * ldc + col];
  return c;
}

// ---------------------------------------------------------------------------
// Async global->LDS copy of one contiguous 2176-byte block (16 rows x 34 f32):
// 4 x b128 (32 lanes x 16B = 512B each) + 1 x b32 tail (32 lanes x 4B = 128B).
// GVS addressing: mem = SGPR64 + VGPR32; LDS dest = per-lane VGPR byte offset.
// Tracked with ASYNCcnt (loads retire in order).
// ---------------------------------------------------------------------------
static __device__ __forceinline__ void async_load_block(const float* gsrc,
                                                        uint32_t lds_base) {
  const uint32_t lane = threadIdx.x & 31;
  const uint64_t g = (uint64_t)(uintptr_t)gsrc;
#pragma unroll
  for (int i = 0; i < 4; ++i) {
    uint32_t voff = lane * 16u + (uint32_t)i * 512u;
    uint32_t dst = lds_base + voff;
    asm volatile("global_load_async_to_lds_b128 %0, %1, %2"
                 :
                 : "v"(dst), "v"(voff), "s"(g)
                 : "memory");
  }
  {
    uint32_t voff = 2048u + lane * 4u;
    uint32_t dst = lds_base + voff;
    asm volatile("global_load_async_to_lds_b32 %0, %1, %2"
                 :
                 : "v"(dst), "v"(voff), "s"(g)
                 : "memory");
  }
}

// ---------------------------------------------------------------------------
// One wave (32 lanes) per batch. Maintains M = (G + eps*I)^-1 in LDS via
// rank-16 Woodbury updates (WMMA f32 16x16x4). Intra-block per-step ATE via
// bordered inverse of the 16x16 capacitance C_k = I + S(1:k,1:k).
//   num(n) = Sty - Xty^T Ginv Xtt,  den(n) = Stt - Xtt^T Ginv Xtt
// Next block's rows are prefetched with async-to-LDS while the serial
// capacitance section runs (single wave per batch -> must self-hide latency).
// ---------------------------------------------------------------------------
__global__ void __launch_bounds__(32)
linear_dml_kernel(const float* __restrict__ in, float* __restrict__ out) {
  __shared__ __align__(16) float Xraw[2][kRow];  // double-buffered raw block
  __shared__ float M[kD * kD];                   // (G + eps I)^-1
  __shared__ float P[kBlk * kD];                 // X * M
  __shared__ float W[kBlk * kD];                 // -Cinv * P
  __shared__ float S[kBlk * kBlk];               // X * M * X^T
  __shared__ float Cinv[kBlk * kBlk];            // (I + S_k)^-1, grown per step
  __shared__ float a0[kD], b0[kD], vb[kD];
  __shared__ float alpha[kBlk], beta[kBlk], hbuf[kBlk], rbuf[kBlk];

  const int lane = threadIdx.x;
  const int b = blockIdx.x;
  const float* batch_in = in + (size_t)b * kN * kC;

  // kick off block 0's async copy immediately
  async_load_block(batch_in, (uint32_t)(uintptr_t)&Xraw[0][0]);

  // ---- init per-batch state (overlaps with async copy) ----
  for (int j = 0; j < kD; ++j)
    M[lane * kD + j] = (lane == j) ? (1.0f / RIDGE_EPS) : 0.0f;
  a0[lane] = 0.0f;
  b0[lane] = 0.0f;
  // second output: 2*log(zeros) == -inf
  const float ninf = -__builtin_inff();
  for (int i = lane; i < kN; i += 32)
    out[(size_t)kB * kN + (size_t)b * kN + i] = ninf;
  float Sty = 0.0f, Stt = 0.0f;
  __syncthreads();

  for (int blk = 0; blk < kNB; ++blk) {
    const int cur = blk & 1;
    const int n0 = blk * kBlk;
    const float* Xc = &Xraw[cur][0];  // row k element j at Xc[k*34 + j]

    // prefetch next block into the other buffer, then wait only for current
    if (blk + 1 < kNB) {
      async_load_block(batch_in + (size_t)(blk + 1) * kRow,
                       (uint32_t)(uintptr_t)&Xraw[cur ^ 1][0]);
      asm volatile("s_wait_asynccnt 0x5" ::: "memory");
    } else {
      asm volatile("s_wait_asynccnt 0x0" ::: "memory");
    }
    __syncthreads();

    // ---- P = X * M (16x32), WMMA ----
    for (int ct = 0; ct < 2; ++ct) {
      v8f acc = {};
      for (int k4 = 0; k4 < 8; ++k4)
        acc = wmma4(&Xc[k4 * 4], kC, &M[(k4 * 4) * kD + ct * 16], kD, acc);
      store_tile(&P[ct * 16], kD, acc, 1.0f);
    }
    __syncthreads();

    // ---- S = P * X^T (16x16), WMMA ----
    {
      v8f acc = {};
      for (int k4 = 0; k4 < 8; ++k4)
        acc = wmma4_bt(&P[k4 * 4], kD, &Xc[k4 * 4], kC, acc);
      store_tile(S, kBlk, acc, 1.0f);
    }
    __syncthreads();

    // ---- block-start vectors: alpha = P*a0, beta = P*b0, vb = M*b0 ----
    if (lane < kBlk) {
      float sa = 0.0f, sb = 0.0f;
      for (int j = 0; j < kD; ++j) {
        sa += P[lane * kD + j] * a0[j];
        sb += P[lane * kD + j] * b0[j];
      }
      alpha[lane] = sa;
      beta[lane] = sb;
    }
    {
      float s = 0.0f;
      for (int j = 0; j < kD; ++j) s += M[lane * kD + j] * b0[j];
      vb[lane] = s;
    }
    for (int i = lane; i < kBlk * kBlk; i += 32) Cinv[i] = 0.0f;
    __syncthreads();

    float m_ab = 0.0f, m_bb = 0.0f;  // a^T M b, b^T M b (wave-uniform)
    for (int j = 0; j < kD; ++j) {
      m_ab += a0[j] * vb[j];
      m_bb += b0[j] * vb[j];
    }

    // ---- 16 sequential steps on the 16x16 capacitance system ----
    float ate_reg = 0.0f;  // lane k keeps step k's (uniform) result
    for (int k = 0; k < kBlk; ++k) {
      // h = Cinv(1:k,1:k) * S(1:k,k)
      if (lane < k) {
        float h = 0.0f;
        for (int j = 0; j < k; ++j) h += Cinv[lane * kBlk + j] * S[j * kBlk + k];
        hbuf[lane] = h;
      }
      __syncthreads();
      float delta = 1.0f + S[k * kBlk + k];
      for (int j = 0; j < k; ++j) delta -= S[j * kBlk + k] * hbuf[j];
      const float inv_d = 1.0f / delta;

      // bordered inverse update of Cinv
      if (lane < k) {
        const float hi = hbuf[lane];
        for (int j = 0; j < k; ++j) Cinv[lane * kBlk + j] += hi * hbuf[j] * inv_d;
        Cinv[lane * kBlk + k] = -hi * inv_d;
      } else if (lane == k) {
        for (int j = 0; j < k; ++j) Cinv[k * kBlk + j] = -hbuf[j] * inv_d;
        Cinv[k * kBlk + k] = inv_d;
      }

      const float tk = Xc[k * kC + kD + 0];
      const float yk = Xc[k * kC + kD + 1];
      const float skk = S[k * kBlk + k];
      m_ab += yk * beta[k] + tk * alpha[k] + yk * tk * skk;
      m_bb += 2.0f * tk * beta[k] + tk * tk * skk;
      Sty += tk * yk;
      Stt += tk * tk;
      __syncthreads();

      if (lane < kBlk) {
        alpha[lane] += yk * S[lane * kBlk + k];
        beta[lane] += tk * S[lane * kBlk + k];
      }
      __syncthreads();

      // r = Cinv(1:k+1,1:k+1) * beta(1:k+1)
      if (lane <= k) {
        float r = 0.0f;
        for (int j = 0; j <= k; ++j) r += Cinv[lane * kBlk + j] * beta[j];
        rbuf[lane] = r;
      }
      __syncthreads();

      float qa = 0.0f, qb = 0.0f;
      for (int i = 0; i <= k; ++i) {
        qa += alpha[i] * rbuf[i];
        qb += beta[i] * rbuf[i];
      }
      const float q1 = m_ab - qa, q2 = m_bb - qb;
      const float num = Sty - q1, den = Stt - q2;
      float ate = (den > 0.0f) ? (num / den) : 0.0f;
      if (n0 + k == 0) ate = 0.0f;  // means[:,0] = 0
      if (lane == k) ate_reg = ate;
      __syncthreads();
    }
    if (lane < kBlk) out[(size_t)b * kN + n0 + lane] = ate_reg;

    // ---- accumulate Xty/Xtt base vectors ----
    {
      float sa = 0.0f, sb = 0.0f;
      for (int k = 0; k < kBlk; ++k) {
        sa += Xc[k * kC + kD + 1] * Xc[k * kC + lane];
        sb += Xc[k * kC + kD + 0] * Xc[k * kC + lane];
      }
      a0[lane] += sa;
      b0[lane] += sb;
    }
    __syncthreads();

    // ---- W = -(Cinv * P) (16x32), WMMA ----
    for (int ct = 0; ct < 2; ++ct) {
      v8f acc = {};
      for (int k4 = 0; k4 < 4; ++k4)
        acc = wmma4(&Cinv[k4 * 4], kBlk, &P[(k4 * 4) * kD + ct * 16], kD, acc);
      store_tile(&W[ct * 16], kD, acc, -1.0f);
    }
    __syncthreads();

    // ---- M += P^T * W  (i.e. M -= P^T Cinv P), WMMA ----
    for (int rt = 0; rt < 2; ++rt)
      for (int ct = 0; ct < 2; ++ct) {
        v8f acc = load_tile(&M[(rt * 16) * kD + ct * 16], kD);
        for (int k4 = 0; k4 < 4; ++k4)
          acc = wmma4_at(&P[(k4 * 4) * kD + rt * 16], kD,
                         &W[(k4 * 4) * kD + ct * 16], kD, acc);
        store_tile(&M[(rt * 16) * kD + ct * 16], kD, acc, 1.0f);
      }
    __syncthreads();
  }
}

extern "C" void kernel_launch(void* const* d_in, const int* in_sizes, int n_in,
                              void* d_out, int out_size, void* d_ws,
                              size_t ws_size, hipStream_t stream) {
  (void)in_sizes; (void)n_in; (void)d_ws; (void)ws_size; (void)out_size;
  const float* xtys = (const float*)d_in[0];
  float* out = (float*)d_out;
  linear_dml_kernel<<<dim3(kB), dim3(32), 0, stream>>>(xtys, out);
}